// S6Block_55190329754137
// MI455X (gfx1250) — compile-verified
//
#include <hip/hip_runtime.h>

typedef float v2f __attribute__((ext_vector_type(2)));
typedef float v8f __attribute__((ext_vector_type(8)));

#define DIM     256
#define BATCH   16
#define SEQ     8192
#define CHUNK   128                 // T: steps per chunk
#define NCHUNK  (SEQ / CHUNK)       // 64 chunks
#define KSTEPS  (DIM / 4)           // 64 WMMA K-steps per [16,256]@[256,256]
#define LDSPITCH 260                // padded row pitch (floats) -> conflict-free b64 reads

// ---------------------------------------------------------------------------
// One chunk of the scan. Each workgroup = 1 chunk, 16 waves (512 threads),
// wave w owns output columns [16w, 16w+16) of the [16,256] state.
// hinit : entry state per chunk (or null -> zeros)
// vout  : write chunk-final state (pass 1)
// y     : emit outputs (pass 2)
// ---------------------------------------------------------------------------
__global__ __launch_bounds__(512)
void s6_chunk_kernel(const float* __restrict__ x,
                     const float* __restrict__ A,
                     const float* __restrict__ Bp,
                     const float* __restrict__ Cp,
                     const float* __restrict__ Dp,
                     const float* __restrict__ rs,
                     const float* __restrict__ hinit,
                     float* __restrict__ vout,
                     float* __restrict__ y)
{
    __shared__ float hbuf[2][BATCH][LDSPITCH];

    const int tid  = threadIdx.x;
    const int wave = tid >> 5;
    const int lane = tid & 31;
    const int l16  = lane & 15;
    const int hi   = lane >> 4;          // half-wave select
    const int colw = wave * 16 + l16;    // global column this lane covers
    const int chunk = blockIdx.x;

    // Stationary right operand: column panel A[:, 16w:16w+16] in B-fragment
    // layout. VGPR0: K rows 4kk+{0|2}; VGPR1: K rows 4kk+{1|3}. 128 VGPRs.
    float bf0[KSTEPS], bf1[KSTEPS];
#pragma unroll
    for (int kk = 0; kk < KSTEPS; ++kk) {
        const int r = 4 * kk + hi * 2;
        bf0[kk] = A[r * DIM + colw];
        bf1[kk] = A[(r + 1) * DIM + colw];
    }

    const float bcol  = Bp[colw];
    const float ccol  = Cp[colw];
    const float drcol = Dp[colw] + rs[0];

    // Entry state -> LDS buffer 0
    for (int idx = tid; idx < BATCH * DIM; idx += 512) {
        const int r = idx >> 8, c = idx & 255;
        hbuf[0][r][c] = hinit ? hinit[chunk * BATCH * DIM + idx] : 0.0f;
    }
    __syncthreads();

    const int t0 = chunk * CHUNK;
    int p = 0;
    v8f acc = {};

#pragma unroll 1
    for (int t = 0; t < CHUNK; ++t) {
        const int xbase = (t0 + t) * DIM + colw;

        // Load x tile in C/D layout and fold B⊙x into the accumulator init.
        float xv[8];
        v8f c;
#pragma unroll
        for (int v = 0; v < 8; ++v) {
            xv[v] = x[(v + hi * 8) * (SEQ * DIM) + xbase];
            c[v]  = bcol * xv[v];
        }

        // h_new tile = h_old @ A[:,panel] + B⊙x : 64 chained fp32 WMMAs.
#pragma unroll
        for (int kk = 0; kk < KSTEPS; ++kk) {
            v2f a = *(const v2f*)&hbuf[p][l16][4 * kk + hi * 2];   // A-frag from LDS
            v2f b = { bf0[kk], bf1[kk] };                          // panel in regs
            c = __builtin_amdgcn_wmma_f32_16x16x4_f32(
                    false, a, false, b, (short)0, c, false, false);
        }

        if (y) {
#pragma unroll
            for (int v = 0; v < 8; ++v)
                y[(v + hi * 8) * (SEQ * DIM) + xbase] = ccol * c[v] + drcol * xv[v];
        }

        // Publish h_new into the other LDS buffer (conflict-free pattern).
#pragma unroll
        for (int v = 0; v < 8; ++v)
            hbuf[p ^ 1][v + hi * 8][colw] = c[v];
        __syncthreads();
        p ^= 1;
        acc = c;
    }

    if (vout) {
#pragma unroll
        for (int v = 0; v < 8; ++v)
            vout[chunk * BATCH * DIM + (v + hi * 8) * DIM + colw] = acc[v];
    }
}

// ---------------------------------------------------------------------------
// 256x256 fp32 matmul (used to build A^CHUNK by repeated squaring).
// 16 blocks x 16 waves = 256 independent 16x16 output tiles.
// ---------------------------------------------------------------------------
__global__ __launch_bounds__(512)
void mm256_kernel(const float* __restrict__ In, float* __restrict__ Out)
{
    const int tid  = threadIdx.x;
    const int wave = tid >> 5;
    const int lane = tid & 31;
    const int l16  = lane & 15;
    const int hi   = lane >> 4;
    const int tile = blockIdx.x * 16 + wave;
    const int m0   = (tile >> 4) * 16;
    const int n0   = (tile & 15) * 16;

    v8f c = {};
#pragma unroll 4
    for (int kk = 0; kk < KSTEPS; ++kk) {
        const int kb = 4 * kk + hi * 2;
        v2f a = { In[(m0 + l16) * DIM + kb], In[(m0 + l16) * DIM + kb + 1] };
        v2f b = { In[kb * DIM + n0 + l16],  In[(kb + 1) * DIM + n0 + l16] };
        c = __builtin_amdgcn_wmma_f32_16x16x4_f32(
                false, a, false, b, (short)0, c, false, false);
    }
#pragma unroll
    for (int v = 0; v < 8; ++v)
        Out[(m0 + v + hi * 8) * DIM + n0 + l16] = c[v];
}

// ---------------------------------------------------------------------------
// Inter-chunk scan: H_0 = 0 ; H_{j+1} = H_j @ A^CHUNK + v_j.
// Writes entry[j] = state entering chunk j. Single WG, 16 waves.
// ---------------------------------------------------------------------------
__global__ __launch_bounds__(512)
void s6_scan_kernel(const float* __restrict__ Apow,   // A^CHUNK [256,256]
                    const float* __restrict__ vst,    // [NCHUNK][16][256]
                    float* __restrict__ entry)        // [NCHUNK][16][256]
{
    __shared__ float hbuf[2][BATCH][LDSPITCH];

    const int tid  = threadIdx.x;
    const int wave = tid >> 5;
    const int lane = tid & 31;
    const int l16  = lane & 15;
    const int hi   = lane >> 4;
    const int colw = wave * 16 + l16;

    float bf0[KSTEPS], bf1[KSTEPS];
#pragma unroll
    for (int kk = 0; kk < KSTEPS; ++kk) {
        const int r = 4 * kk + hi * 2;
        bf0[kk] = Apow[r * DIM + colw];
        bf1[kk] = Apow[(r + 1) * DIM + colw];
    }

    for (int idx = tid; idx < BATCH * DIM; idx += 512) {
        entry[idx] = 0.0f;                              // H_0 = 0
        hbuf[0][idx >> 8][idx & 255] = 0.0f;
    }
    __syncthreads();

    int p = 0;
#pragma unroll 1
    for (int j = 0; j < NCHUNK; ++j) {
        v8f c;
#pragma unroll
        for (int v = 0; v < 8; ++v)
            c[v] = vst[j * BATCH * DIM + (v + hi * 8) * DIM + colw];

#pragma unroll
        for (int kk = 0; kk < KSTEPS; ++kk) {
            v2f a = *(const v2f*)&hbuf[p][l16][4 * kk + hi * 2];
            v2f b = { bf0[kk], bf1[kk] };
            c = __builtin_amdgcn_wmma_f32_16x16x4_f32(
                    false, a, false, b, (short)0, c, false, false);
        }

        if (j + 1 < NCHUNK) {
#pragma unroll
            for (int v = 0; v < 8; ++v)
                entry[(j + 1) * BATCH * DIM + (v + hi * 8) * DIM + colw] = c[v];
        }
#pragma unroll
        for (int v = 0; v < 8; ++v)
            hbuf[p ^ 1][v + hi * 8][colw] = c[v];
        __syncthreads();
        p ^= 1;
    }
}

// ---------------------------------------------------------------------------
extern "C" void kernel_launch(void* const* d_in, const int* in_sizes, int n_in,
                              void* d_out, int out_size, void* d_ws, size_t ws_size,
                              hipStream_t stream)
{
    const float* x  = (const float*)d_in[0];
    const float* A  = (const float*)d_in[1];
    const float* Bp = (const float*)d_in[2];
    const float* Cp = (const float*)d_in[3];
    const float* Dp = (const float*)d_in[4];
    const float* rs = (const float*)d_in[5];
    float* y  = (float*)d_out;
    float* ws = (float*)d_ws;

    float* vst   = ws;                               // NCHUNK*16*256  (1 MB)
    float* entry = vst + NCHUNK * BATCH * DIM;       // NCHUNK*16*256  (1 MB)
    float* pw0   = entry + NCHUNK * BATCH * DIM;     // 256*256        (256 KB)
    float* pw1   = pw0 + DIM * DIM;                  // 256*256        (256 KB)

    const dim3 blk(512);

    // Pass 1: per-chunk local scans from zero -> chunk contributions v_j.
    s6_chunk_kernel<<<NCHUNK, blk, 0, stream>>>(x, A, Bp, Cp, Dp, rs,
                                                nullptr, vst, nullptr);

    // A^128 by 7 squarings (CHUNK = 128). Result ends in pw0.
    mm256_kernel<<<16, blk, 0, stream>>>(A,   pw0);  // A^2
    mm256_kernel<<<16, blk, 0, stream>>>(pw0, pw1);  // A^4
    mm256_kernel<<<16, blk, 0, stream>>>(pw1, pw0);  // A^8
    mm256_kernel<<<16, blk, 0, stream>>>(pw0, pw1);  // A^16
    mm256_kernel<<<16, blk, 0, stream>>>(pw1, pw0);  // A^32
    mm256_kernel<<<16, blk, 0, stream>>>(pw0, pw1);  // A^64
    mm256_kernel<<<16, blk, 0, stream>>>(pw1, pw0);  // A^128

    // Inter-chunk scan: entry states H_j.
    s6_scan_kernel<<<1, blk, 0, stream>>>(pw0, vst, entry);

    // Pass 2: re-run chunks from true entry states, emit y.
    s6_chunk_kernel<<<NCHUNK, blk, 0, stream>>>(x, A, Bp, Cp, Dp, rs,
                                                entry, nullptr, y);
}